// LossFunc_13752485282042
// MI455X (gfx1250) — compile-verified
//
#include <hip/hip_runtime.h>
#include <hip/hip_bf16.h>

typedef __attribute__((ext_vector_type(2))) float v2f;
typedef __attribute__((ext_vector_type(8))) float v8f;
typedef int v4i __attribute__((vector_size(16)));

#define BB   256
#define NN   512
#define KK   4
#define DD   9
#define LL   32

#define GLOBAL_AS __attribute__((address_space(1)))
#define LDS_AS    __attribute__((address_space(3)))

#if defined(__has_builtin)
#if __has_builtin(__builtin_amdgcn_global_load_async_to_lds_b128)
#define HAVE_ASYNC_LDS 1
#endif
#endif
#ifndef HAVE_ASYNC_LDS
#define HAVE_ASYNC_LDS 0
#endif

__global__ __launch_bounds__(256)
void loss_kernel(const float* __restrict__ kine_input,
                 const float* __restrict__ class_input,
                 const float* __restrict__ kine_pred,
                 const float* __restrict__ class_pred,
                 const float* __restrict__ mu,
                 const float* __restrict__ log_var,
                 float* __restrict__ out)
{
    __shared__ float xs[NN * KK];                 // 8 KB  kine_input[b]
    __shared__ float ys[NN * KK];                 // 8 KB  kine_pred[b]
    __shared__ float x2s[NN];                     // 2 KB  row squared norms
    __shared__ float y2s[NN];                     // 2 KB  col squared norms
    __shared__ unsigned long long row_min[NN];    // 4 KB  packed (dist1, idx1)
    __shared__ unsigned long long col_min[NN];    // 4 KB  packed (dist2, idx2)
    __shared__ int   cnt_in[DD];
    __shared__ int   cnt_pred[DD];
    __shared__ float acc_chamfer, acc_class, acc_kl;

    const int b    = blockIdx.x;
    const int t    = threadIdx.x;     // 0..255  (8 waves of 32)
    const int lane = t & 31;
    const int wave = t >> 5;

    const float* xg  = kine_input  + (size_t)b * NN * KK;
    const float* yg  = kine_pred   + (size_t)b * NN * KK;
    const float* cig = class_input + (size_t)b * NN * DD;
    const float* cpg = class_pred  + (size_t)b * NN * DD;

    // ---------------- Phase 0: stage point clouds + init ----------------
#if HAVE_ASYNC_LDS
    // gfx1250 async copy: global -> LDS without touching VGPRs (ASYNCcnt).
    for (int i = t; i < NN; i += 256) {
        __builtin_amdgcn_global_load_async_to_lds_b128(
            (GLOBAL_AS v4i*)(xg + (size_t)i * 4),
            (LDS_AS   v4i*)(xs + (size_t)i * 4), 0, 0);
        __builtin_amdgcn_global_load_async_to_lds_b128(
            (GLOBAL_AS v4i*)(yg + (size_t)i * 4),
            (LDS_AS   v4i*)(ys + (size_t)i * 4), 0, 0);
        row_min[i] = 0xFFFFFFFFFFFFFFFFull;
        col_min[i] = 0xFFFFFFFFFFFFFFFFull;
    }
#else
    for (int i = t; i < NN; i += 256) {
        float4 vx = ((const float4*)xg)[i];
        float4 vy = ((const float4*)yg)[i];
        ((float4*)xs)[i] = vx;
        ((float4*)ys)[i] = vy;
        row_min[i] = 0xFFFFFFFFFFFFFFFFull;
        col_min[i] = 0xFFFFFFFFFFFFFFFFull;
    }
#endif
    if (t < DD) { cnt_in[t] = 0; cnt_pred[t] = 0; }
    if (t == 0) { acc_chamfer = 0.0f; acc_class = 0.0f; acc_kl = 0.0f; }

    // Prefetch Phase-2 class data into cache while the WMMA phase runs.
    // 18.4 KB per tensor per block; one 128B line per thread (t < 144).
    if (t < 144) {
        __builtin_prefetch(cig + (size_t)t * 32, 0, 0);
        __builtin_prefetch(cpg + (size_t)t * 32, 0, 0);
    }

#if HAVE_ASYNC_LDS
    asm volatile("s_wait_asynccnt 0x0" ::: "memory");
#endif
    __syncthreads();

    // squared norms from staged LDS data
    for (int i = t; i < NN; i += 256) {
        float4 vx = ((const float4*)xs)[i];
        float4 vy = ((const float4*)ys)[i];
        x2s[i] = vx.x * vx.x + vx.y * vx.y + vx.z * vx.z + vx.w * vx.w;
        y2s[i] = vy.x * vy.x + vy.y * vy.y + vy.z * vy.z + vy.w * vy.w;
    }
    __syncthreads();

    // -------- Phase 1: pairwise sqdist tiles via V_WMMA_F32_16X16X4_F32 --------
    // A (16x4) element (m,k): lane = (k>=2)*16 + m, vgpr = k&1
    // B (4x16) element (k,n): lane = (k>=2)*16 + n, vgpr = k&1
    // C/D (16x16): lane = half*16 + n(idx), vgpr i -> row M = i + 8*half
    const int idx  = lane & 15;
    const int half = lane >> 4;

    for (int rt = wave; rt < NN / 16; rt += 8) {
        const int rowbase = rt * 16;
        const int m = rowbase + idx;
        v2f a;
        a.x = xs[m * KK + 2 * half + 0];
        a.y = xs[m * KK + 2 * half + 1];

        float runV[8];
        int   runC[8];
#pragma unroll
        for (int i = 0; i < 8; ++i) { runV[i] = 3.0e38f; runC[i] = 0; }

        for (int ct = 0; ct < NN / 16; ++ct) {
            const int colbase = ct * 16;
            const int n = colbase + idx;
            v2f bv;
            bv.x = ys[n * KK + 2 * half + 0];
            bv.y = ys[n * KK + 2 * half + 1];

            v8f c = {};
            // D = A x B (+0):   x[16x4] . y^T[4x16]  -> 16x16 inner products
            c = __builtin_amdgcn_wmma_f32_16x16x4_f32(
                    /*neg_a=*/false, a, /*neg_b=*/false, bv,
                    /*c_mod=*/(short)0, c, /*reuse_a=*/false, /*reuse_b=*/false);

            const float y2v = y2s[n];
            float cminV = 3.0e38f;
            int   cminR = 0;
#pragma unroll
            for (int i = 0; i < 8; ++i) {
                const int r = rowbase + i + 8 * half;
                float v = fmaxf(x2s[r] + y2v - 2.0f * c[i], 0.0f);
                if (v < runV[i]) { runV[i] = v; runC[i] = n; }   // row running min
                if (v < cminV)  { cminV  = v; cminR  = r; }      // col partial min
            }
            // one packed (dist,rowidx) min per lane per column tile
            unsigned long long ckey =
                ((unsigned long long)__float_as_uint(cminV) << 32) | (unsigned)cminR;
            atomicMin(&col_min[n], ckey);
        }
        // commit row running mins (packed dist,colidx)
#pragma unroll
        for (int i = 0; i < 8; ++i) {
            const int r = rowbase + i + 8 * half;
            unsigned long long rkey =
                ((unsigned long long)__float_as_uint(runV[i]) << 32) | (unsigned)runC[i];
            atomicMin(&row_min[r], rkey);
        }
    }
    __syncthreads();

    // -------- Phase 2: chamfer sums, class gathers, histograms, KL --------
    float my_ch = 0.0f, my_cl = 0.0f;
    for (int n = t; n < NN; n += 256) {
        unsigned long long k1 = row_min[n];
        float d1 = __uint_as_float((unsigned)(k1 >> 32));
        int   i1 = (int)(k1 & 0xFFFFFFFFu);
        unsigned long long k2 = col_min[n];
        float d2 = __uint_as_float((unsigned)(k2 >> 32));
        int   i2 = (int)(k2 & 0xFFFFFFFFu);
        my_ch += d1 + d2;

        float s = 0.0f;
        float ci_best = -3.0e38f; int ci_arg = 0;
        float cp_best = -3.0e38f; int cp_arg = 0;
#pragma unroll
        for (int dd = 0; dd < DD; ++dd) {
            float ci = cig[n * DD + dd];
            float cp = cpg[n * DD + dd];
            s += cpg[i1 * DD + dd] * ci;   // sorted_pred * class_input
            s += cig[i2 * DD + dd] * cp;   // sorted_input * class_pred
            if (ci > ci_best) { ci_best = ci; ci_arg = dd; }
            if (cp > cp_best) { cp_best = cp; cp_arg = dd; } // argmax(exp(x)) == argmax(x)
        }
        my_cl += s;
        atomicAdd(&cnt_in[ci_arg], 1);
        atomicAdd(&cnt_pred[cp_arg], 1);
    }
    atomicAdd(&acc_chamfer, my_ch);
    atomicAdd(&acc_class,   my_cl);

    if (t < LL) {
        float m_ = mu[(size_t)b * LL + t];
        float lv = log_var[(size_t)b * LL + t];
        atomicAdd(&acc_kl, 1.0f + lv - m_ * m_ - expf(lv));
    }
    __syncthreads();

    // -------- Phase 3: combine --------
    if (t == 0) {
        float classnum = 0.0f;
#pragma unroll
        for (int dd = 0; dd < DD; ++dd) {
            float diff = fabsf((float)(cnt_pred[dd] - cnt_in[dd]));
            float w = (dd == 0) ? 2.0f : ((dd == DD - 1) ? 100.0f : 1.0f);
            classnum += w * diff;
        }
        const float BETA = 0.01f, W = 1.0f, C = 0.001f;
        float chamfer_loss = acc_chamfer;
        float class_loss   = -acc_class;
        float kl_loss      = -0.5f * acc_kl;
        out[b] = (1.0f - BETA) * (chamfer_loss + W * class_loss + C * classnum)
               + BETA * kl_loss;
    }
}

extern "C" void kernel_launch(void* const* d_in, const int* in_sizes, int n_in,
                              void* d_out, int out_size, void* d_ws, size_t ws_size,
                              hipStream_t stream) {
    const float* kine_input  = (const float*)d_in[0];
    const float* class_input = (const float*)d_in[1];
    const float* kine_pred   = (const float*)d_in[2];
    const float* class_pred  = (const float*)d_in[3];
    const float* mu          = (const float*)d_in[4];
    const float* log_var     = (const float*)d_in[5];
    float* out = (float*)d_out;

    (void)in_sizes; (void)n_in; (void)out_size; (void)d_ws; (void)ws_size;

    loss_kernel<<<BB, 256, 0, stream>>>(kine_input, class_input, kine_pred,
                                        class_pred, mu, log_var, out);
}